// GaussianProcess_72782515798152
// MI455X (gfx1250) — compile-verified
//
#include <hip/hip_runtime.h>
#include <math.h>

// CDNA5 / gfx1250: wave32. WMMA f32 16x16x4 used for the rank-1 S update.
typedef __attribute__((ext_vector_type(2))) float v2f;
typedef __attribute__((ext_vector_type(8))) float v8f;

#ifndef GP_TWO_PI
#define GP_TWO_PI 6.283185307179586476925286766559
#endif

// ---------------------------------------------------------------------------
// Kernel A: celerite factorization (sequential scan, 1 wave).
// S held in WMMA C-layout: s[v] in lane l = S[v + 8*(l>>4)][l&15].
// ---------------------------------------------------------------------------
__global__ __launch_bounds__(32)
void gp_factor_kernel(const float* __restrict__ a, const float* __restrict__ U,
                      const float* __restrict__ V, const float* __restrict__ P,
                      float* __restrict__ dvec, float* __restrict__ Wmat,
                      double* __restrict__ sumlogd, int N)
{
    const int l = threadIdx.x;     // 0..31
    const int c = l & 15;          // column index (and A/B row index for K=0 slot)
    const int h = l >> 4;          // half of wave: rows 0-7 (h=0) / 8-15 (h=1)

    // ---- n = 0 ----
    const float d0 = a[0];
    float w = V[c] / d0;           // W0[c]
    v8f s = {0.f, 0.f, 0.f, 0.f, 0.f, 0.f, 0.f, 0.f};
    v2f aa, bb;
    aa.x = (h == 0) ? d0 * w : 0.0f;  aa.y = 0.0f;   // A: col K=0 = d0*W0
    bb.x = (h == 0) ? w : 0.0f;       bb.y = 0.0f;   // B: row K=0 = W0^T
    // S = d0 * outer(W0, W0)
    s = __builtin_amdgcn_wmma_f32_16x16x4_f32(false, aa, false, bb,
                                              (short)0, s, false, false);
    if (l == 0) dvec[0] = d0;
    if (h == 0) Wmat[c] = w;

    double acc = (double)logf(d0);

    // ---- n = 1 .. N-1 ----
    for (int n = 1; n < N; ++n) {
        const float u  = U[n * 16 + c];
        const float vv = V[n * 16 + c];
        const float p  = P[(n - 1) * 16 + c];
        const float an = a[n];

        if (n + 8 < N) {
            __builtin_prefetch(&U[(n + 8) * 16], 0, 1);
            __builtin_prefetch(&V[(n + 8) * 16], 0, 1);
            __builtin_prefetch(&P[(n + 7) * 16], 0, 1);
        }

        // Sn = (p p^T) o S, and partial of SU[c] = sum_m S[m][c]*U[m] (S symmetric)
        float part = 0.0f;
#pragma unroll
        for (int v = 0; v < 8; ++v) {
            const int row = v + 8 * h;
            const float prow = __shfl(p, row, 32);   // P[row]
            const float urow = __shfl(u, row, 32);   // U[row]
            const float sv = s[v] * (prow * p);
            s[v] = sv;
            part = fmaf(sv, urow, part);
        }
        // SU[c]: sum the two row-halves
        float su = part + __shfl_xor(part, 16, 32);

        // dn = an - sum_c U[c]*SU[c]  (reduce over 16 lanes of each half)
        float t = u * su;
        t += __shfl_xor(t, 1, 32);
        t += __shfl_xor(t, 2, 32);
        t += __shfl_xor(t, 4, 32);
        t += __shfl_xor(t, 8, 32);
        const float dn = an - t;
        const float wn = (vv - su) / dn;

        // S += dn * outer(Wn, Wn) via WMMA rank-1 (only K=0 slot populated)
        aa.x = (h == 0) ? dn * wn : 0.0f;
        bb.x = (h == 0) ? wn : 0.0f;
        s = __builtin_amdgcn_wmma_f32_16x16x4_f32(false, aa, false, bb,
                                                  (short)0, s, false, false);

        if (l == 0) dvec[n] = dn;
        if (h == 0) Wmat[n * 16 + c] = wn;
        acc += (double)logf(dn);
    }
    if (l == 0) *sumlogd = acc;
}

// ---------------------------------------------------------------------------
// Kernel B: forward (lower) substitution, 1 wave. State F[16][4]:
// lane l holds F[l&15][2*(l>>4) + {0,1}]. Carry is the *raw* z; store z/d.
// ---------------------------------------------------------------------------
__global__ __launch_bounds__(32)
void gp_forward_kernel(const float* __restrict__ U, const float* __restrict__ P,
                       const float* __restrict__ y, const float* __restrict__ dvec,
                       const float* __restrict__ Wmat, float* __restrict__ zdiv,
                       int N)
{
    const int l = threadIdx.x;
    const int j = l & 15;
    const int h = l >> 4;

    float f0 = 0.0f, f1 = 0.0f;
    float zp0 = y[2 * h + 0];     // raw z[0]
    float zp1 = y[2 * h + 1];
    {
        const float dinv = 1.0f / dvec[0];
        if (j == 0) {
            zdiv[2 * h + 0] = zp0 * dinv;
            zdiv[2 * h + 1] = zp1 * dinv;
        }
    }

    for (int n = 1; n < N; ++n) {
        const float wp = Wmat[(n - 1) * 16 + j];
        const float pp = P[(n - 1) * 16 + j];
        const float u  = U[n * 16 + j];
        const float y0 = y[n * 4 + 2 * h + 0];
        const float y1 = y[n * 4 + 2 * h + 1];

        if (n + 8 < N) {
            __builtin_prefetch(&Wmat[(n + 7) * 16], 0, 1);
            __builtin_prefetch(&U[(n + 8) * 16], 0, 1);
            __builtin_prefetch(&P[(n + 7) * 16], 0, 1);
            __builtin_prefetch(&y[(n + 8) * 4], 0, 1);
        }

        f0 = pp * (f0 + wp * zp0);
        f1 = pp * (f1 + wp * zp1);

        float t0 = u * f0, t1 = u * f1;
        t0 += __shfl_xor(t0, 1, 32);  t1 += __shfl_xor(t1, 1, 32);
        t0 += __shfl_xor(t0, 2, 32);  t1 += __shfl_xor(t1, 2, 32);
        t0 += __shfl_xor(t0, 4, 32);  t1 += __shfl_xor(t1, 4, 32);
        t0 += __shfl_xor(t0, 8, 32);  t1 += __shfl_xor(t1, 8, 32);

        const float zn0 = y0 - t0;
        const float zn1 = y1 - t1;
        const float dinv = 1.0f / dvec[n];
        if (j == 0) {
            zdiv[n * 4 + 2 * h + 0] = zn0 * dinv;
            zdiv[n * 4 + 2 * h + 1] = zn1 * dinv;
        }
        zp0 = zn0;
        zp1 = zn1;
    }
}

// ---------------------------------------------------------------------------
// Kernel C: backward (upper) substitution + log-likelihood, 1 wave.
// ---------------------------------------------------------------------------
__global__ __launch_bounds__(32)
void gp_backward_kernel(const float* __restrict__ U, const float* __restrict__ P,
                        const float* __restrict__ Wmat, const float* __restrict__ zdiv,
                        const float* __restrict__ y, const double* __restrict__ sumlogd,
                        float* __restrict__ x, float* __restrict__ loglike, int N)
{
    const int l = threadIdx.x;
    const int j = l & 15;
    const int h = l >> 4;

    float g0 = 0.0f, g1 = 0.0f;

    // x[N-1] = z[N-1] (divided)
    const float xl0 = zdiv[(N - 1) * 4 + 2 * h + 0];
    const float xl1 = zdiv[(N - 1) * 4 + 2 * h + 1];
    if (j == 0) {
        x[(N - 1) * 4 + 2 * h + 0] = xl0;
        x[(N - 1) * 4 + 2 * h + 1] = xl1;
    }
    double yd0 = (double)y[(N - 1) * 4 + 2 * h + 0] * (double)xl0;
    double yd1 = (double)y[(N - 1) * 4 + 2 * h + 1] * (double)xl1;

    for (int n = N - 2; n >= 0; --n) {
        const float u1  = U[(n + 1) * 16 + j];
        const float zd0 = zdiv[(n + 1) * 4 + 2 * h + 0];
        const float zd1 = zdiv[(n + 1) * 4 + 2 * h + 1];
        const float pn  = P[n * 16 + j];
        const float wn  = Wmat[n * 16 + j];
        const float zn0 = zdiv[n * 4 + 2 * h + 0];
        const float zn1 = zdiv[n * 4 + 2 * h + 1];
        const float yy0 = y[n * 4 + 2 * h + 0];
        const float yy1 = y[n * 4 + 2 * h + 1];

        if (n >= 8) {
            __builtin_prefetch(&U[(n - 7) * 16], 0, 1);
            __builtin_prefetch(&P[(n - 8) * 16], 0, 1);
            __builtin_prefetch(&Wmat[(n - 8) * 16], 0, 1);
            __builtin_prefetch(&zdiv[(n - 8) * 4], 0, 1);
            __builtin_prefetch(&y[(n - 8) * 4], 0, 1);
        }

        g0 = pn * (g0 + u1 * zd0);
        g1 = pn * (g1 + u1 * zd1);

        float t0 = wn * g0, t1 = wn * g1;
        t0 += __shfl_xor(t0, 1, 32);  t1 += __shfl_xor(t1, 1, 32);
        t0 += __shfl_xor(t0, 2, 32);  t1 += __shfl_xor(t1, 2, 32);
        t0 += __shfl_xor(t0, 4, 32);  t1 += __shfl_xor(t1, 4, 32);
        t0 += __shfl_xor(t0, 8, 32);  t1 += __shfl_xor(t1, 8, 32);

        const float xn0 = zn0 - t0;
        const float xn1 = zn1 - t1;
        if (j == 0) {
            x[n * 4 + 2 * h + 0] = xn0;
            x[n * 4 + 2 * h + 1] = xn1;
        }
        yd0 += (double)yy0 * (double)xn0;
        yd1 += (double)yy1 * (double)xn1;
    }

    if (j == 0) {
        const double norm = -0.5 * (*sumlogd + (double)N * log(GP_TWO_PI));
        loglike[2 * h + 0] = (float)(norm - 0.5 * yd0);
        loglike[2 * h + 1] = (float)(norm - 0.5 * yd1);
    }
}

// ---------------------------------------------------------------------------
// Launch: factor -> forward -> backward (stream-ordered; graph-capture safe).
// Workspace layout (floats): d[0,N) | W[N,17N) | zdiv[17N,21N) | sumlogd(double)
// ---------------------------------------------------------------------------
extern "C" void kernel_launch(void* const* d_in, const int* in_sizes, int n_in,
                              void* d_out, int out_size, void* d_ws, size_t ws_size,
                              hipStream_t stream)
{
    const float* a = (const float*)d_in[0];
    const float* U = (const float*)d_in[1];
    const float* V = (const float*)d_in[2];
    const float* P = (const float*)d_in[3];
    const float* y = (const float*)d_in[4];
    const int N = in_sizes[0];

    float* ws        = (float*)d_ws;
    float* dvec      = ws;
    float* Wmat      = ws + (size_t)N;
    float* zdiv      = ws + (size_t)17 * N;
    double* sumlogd  = (double*)(ws + (size_t)21 * N);

    float* x       = (float*)d_out;
    float* loglike = x + (size_t)N * 4;

    gp_factor_kernel<<<1, 32, 0, stream>>>(a, U, V, P, dvec, Wmat, sumlogd, N);
    gp_forward_kernel<<<1, 32, 0, stream>>>(U, P, y, dvec, Wmat, zdiv, N);
    gp_backward_kernel<<<1, 32, 0, stream>>>(U, P, Wmat, zdiv, y, sumlogd, x, loglike, N);
}